// SparseBmfrm_62809601737271
// MI455X (gfx1250) — compile-verified
//
#include <hip/hip_runtime.h>
#include <stdint.h>

#ifndef __has_builtin
#define __has_builtin(x) 0
#endif

#define AS1 __attribute__((address_space(1)))
#define AS3 __attribute__((address_space(3)))

typedef int v4i __attribute__((vector_size(16)));

// ---- problem constants (from reference) ----
#define NC_   128
#define NS_   2048
#define NX_   256
#define NZ_   512
#define NSNC_ (NS_ * NC_)   // 262144  (rhs rows)
#define NXNZ_ (NX_ * NZ_)   // 131072  (output rows)

// ---- tiling ----
#define THREADS 256
#define CHUNK   16                  // contiguous nnz per thread (sorted rows => ~1 flush)
#define TILE    (THREADS * CHUNK)   // 4096 nnz per workgroup

#if __has_builtin(__builtin_amdgcn_global_load_async_to_lds_b128)
#define HAVE_ASYNC_BUILTIN 1
#else
#define HAVE_ASYNC_BUILTIN 0
#endif

// Copy 16 bytes global -> LDS asynchronously (tracked by ASYNCcnt).
__device__ __forceinline__ void async_copy16(const void* g, void* l) {
#if HAVE_ASYNC_BUILTIN
  __builtin_amdgcn_global_load_async_to_lds_b128(
      (AS1 v4i*)(uintptr_t)g, (AS3 v4i*)(uintptr_t)l, 0, 0);
#else
  // VDST = LDS byte address (low 32 bits of generic LDS addr), VADDR = 64-bit global addr
  unsigned lds_off = (unsigned)(uintptr_t)l;
  asm volatile("global_load_async_to_lds_b128 %0, %1, off"
               :: "v"(lds_off), "v"(g)
               : "memory");
#endif
}

__device__ __forceinline__ void wait_async0() {
#if __has_builtin(__builtin_amdgcn_s_wait_asynccnt)
  __builtin_amdgcn_s_wait_asynccnt(0);
#else
  asm volatile("s_wait_asynccnt 0" ::: "memory");
#endif
}

__device__ __forceinline__ void atomic_add_f32(float* p, float v) {
  __hip_atomic_fetch_add(p, v, __ATOMIC_RELAXED, __HIP_MEMORY_SCOPE_AGENT);
}

// out[bk][z][x] layout: bk*NXNZ + z*NX + x, with row r = x*NZ + z
__device__ __forceinline__ void flush_row(float* __restrict__ out, int r,
                                          float a0, float a1, float a2, float a3) {
  if (r < 0) return;
  int z = r & (NZ_ - 1);
  int x = r >> 9;                 // NZ_ == 512
  int o = z * NX_ + x;
  atomic_add_f32(out + 0 * NXNZ_ + o, a0);
  atomic_add_f32(out + 1 * NXNZ_ + o, a1);
  atomic_add_f32(out + 2 * NXNZ_ + o, a2);
  atomic_add_f32(out + 3 * NXNZ_ + o, a3);
}

__global__ void zero_kernel(float* __restrict__ out, int n) {
  int i = blockIdx.x * blockDim.x + threadIdx.x;
  if (i < n) out[i] = 0.0f;
}

// rhs4[c] = { x[bk*NSNC + (c%NC)*NS + (c/NC)] for bk in 0..3 }
__global__ void pack_rhs_kernel(const float* __restrict__ x, float4* __restrict__ rhs4) {
  int c = blockIdx.x * blockDim.x + threadIdx.x;
  if (c >= NSNC_) return;
  int cc  = c & (NC_ - 1);
  int s   = c >> 7;               // NC_ == 128
  int off = cc * NS_ + s;
  float4 v;
  v.x = x[0 * NSNC_ + off];
  v.y = x[1 * NSNC_ + off];
  v.z = x[2 * NSNC_ + off];
  v.w = x[3 * NSNC_ + off];
  rhs4[c] = v;
}

template <bool PACKED>
__global__ __launch_bounds__(THREADS) void spmm_kernel(
    const float* __restrict__ vals, const int* __restrict__ rows,
    const int* __restrict__ cols, const float4* __restrict__ rhs4,
    const float* __restrict__ x, float* __restrict__ out, long long nnz) {
  __shared__ float vLds[TILE];   // 16 KB
  __shared__ int   rLds[TILE];   // 16 KB
  __shared__ int   cLds[TILE];   // 16 KB  (48 KB total -> several WGs per 320 KB WGP)

  const int t = threadIdx.x;
  const long long tileBase = (long long)blockIdx.x * TILE;
  const long long base     = tileBase + (long long)t * CHUNK;
  const bool fullTile      = (tileBase + TILE) <= nnz;

  if (fullTile) {
    // Each thread stages its own contiguous 64B chunk of each stream via
    // async b128 copies. Only this wave's own data is consumed, so a
    // per-wave s_wait_asynccnt suffices (no workgroup barrier needed).
    const char* gv = (const char*)(vals + base);
    const char* gr = (const char*)(rows + base);
    const char* gc = (const char*)(cols + base);
    char* lv = (char*)(vLds + t * CHUNK);
    char* lr = (char*)(rLds + t * CHUNK);
    char* lc = (char*)(cLds + t * CHUNK);
#pragma unroll
    for (int i = 0; i < CHUNK * 4; i += 16) {
      async_copy16(gv + i, lv + i);
      async_copy16(gr + i, lr + i);
      async_copy16(gc + i, lc + i);
    }
    wait_async0();
  }

  float a0 = 0.f, a1 = 0.f, a2 = 0.f, a3 = 0.f;
  int curRow = -1;

#pragma unroll 4
  for (int i = 0; i < CHUNK; ++i) {
    float v; int r, c;
    if (fullTile) {
      v = vLds[t * CHUNK + i];
      r = rLds[t * CHUNK + i];
      c = cLds[t * CHUNK + i];
    } else {
      long long g = base + i;
      if (g >= nnz) break;
      v = vals[g]; r = rows[g]; c = cols[g];
    }

    float4 rv;
    if (PACKED) {
      rv = rhs4[c];                       // one 16B L2-resident gather
    } else {
      int off = (c & (NC_ - 1)) * NS_ + (c >> 7);
      rv.x = x[0 * NSNC_ + off];
      rv.y = x[1 * NSNC_ + off];
      rv.z = x[2 * NSNC_ + off];
      rv.w = x[3 * NSNC_ + off];
    }

    if (r != curRow) {                    // sorted rows: rare (avg run length ~128)
      flush_row(out, curRow, a0, a1, a2, a3);
      curRow = r;
      a0 = a1 = a2 = a3 = 0.f;
    }
    a0 = fmaf(v, rv.x, a0);
    a1 = fmaf(v, rv.y, a1);
    a2 = fmaf(v, rv.z, a2);
    a3 = fmaf(v, rv.w, a3);
  }
  flush_row(out, curRow, a0, a1, a2, a3);
}

extern "C" void kernel_launch(void* const* d_in, const int* in_sizes, int n_in,
                              void* d_out, int out_size, void* d_ws, size_t ws_size,
                              hipStream_t stream) {
  const float* x    = (const float*)d_in[0];
  const float* vals = (const float*)d_in[1];
  const int*   rows = (const int*)d_in[2];   // harness: integer -> const int*
  const int*   cols = (const int*)d_in[3];
  float* out = (float*)d_out;
  const long long nnz = (long long)in_sizes[1];

  // 1) zero output (we accumulate with atomics; d_out is poisoned)
  int zb = (out_size + THREADS - 1) / THREADS;
  zero_kernel<<<zb, THREADS, 0, stream>>>(out, out_size);

  // 2) pack rhs gather target into float4s if workspace allows (4 MB)
  const bool packed = ws_size >= (size_t)NSNC_ * sizeof(float4);
  float4* rhs4 = (float4*)d_ws;
  if (packed) {
    pack_rhs_kernel<<<(NSNC_ + THREADS - 1) / THREADS, THREADS, 0, stream>>>(x, rhs4);
  }

  // 3) bandwidth-bound streaming SpMM with async-LDS staging
  int nTiles = (int)((nnz + TILE - 1) / TILE);
  if (nTiles < 1) nTiles = 1;
  if (packed)
    spmm_kernel<true><<<nTiles, THREADS, 0, stream>>>(vals, rows, cols, rhs4, x, out, nnz);
  else
    spmm_kernel<false><<<nTiles, THREADS, 0, stream>>>(vals, rows, cols, rhs4, x, out, nnz);
}